// BiAttn_TFN_hg_2desc_Net_84954453115068
// MI455X (gfx1250) — compile-verified
//
#include <hip/hip_runtime.h>
#include <hip/hip_bf16.h>
#include <math.h>

// ---------------------------------------------------------------------------
// Sizes (from reference)
// ---------------------------------------------------------------------------
#define N_NODES 50000
#define N_EDGES 800000
#define BATCH   512
#define DIM_IN  128
#define D2      200
#define GC1     100
#define DG      20
#define DH      64
#define H1      128
#define H2      32
#define FUSE    4221        // (DG+1)*(D2+1)
#define FUSEP   4224        // padded to multiple of 4 for WMMA K
#define GC1P    112         // GC1 padded to 7*16 tiles
#define DGP     32          // DG padded to 2*16 tiles
#define BN_EPS  1e-5f

typedef __attribute__((ext_vector_type(2))) float v2f;
typedef __attribute__((ext_vector_type(8))) float v8f;

// ---------------------------------------------------------------------------
// Packed fp32 WMMA GEMM:  C[M, NT*16] = A[M,K] @ W[K, NT*16] (+bias)
//
// One wave per 16-row strip; the wave holds NT 16x16 accumulator tiles and
// reuses each A fragment across all NT WMMAs. B comes from a pre-packed
// buffer Wp[pair][col][2] (pair = k/2, col padded to NT*16, zeros in pad),
// so every B load is one unconditional, coalesced global_load_b64.
//
// Fragment layouts per ISA 7.12.2 (32-bit, wave32):
//   A 16x4 : lanes 0-15 K={0,1}, lanes 16-31 K={2,3}, M = lane&15
//   B 4x16 : mirrored, N = lane&15
//   C/D    : VGPR r -> row = r + 8*(lane>=16), col = lane&15
// ---------------------------------------------------------------------------
template <int NT>
__global__ __launch_bounds__(128)
void wmma_gemm_f32(const float* __restrict__ A, int lda,
                   const float* __restrict__ Wp,      // packed [K/2][NT*16][2]
                   float* __restrict__ C, int ldc,
                   const float* __restrict__ bias,    // NT*16 entries or null
                   int mtiles, int K)
{
    const int ncolsP = NT * 16;
    int wave = (blockIdx.x * blockDim.x + threadIdx.x) >> 5;
    int lane = threadIdx.x & 31;
    if (wave >= mtiles) return;                  // wave-uniform guard
    int m0   = wave << 4;
    int half = lane >> 4;                        // 0: lanes 0-15, 1: lanes 16-31
    int l    = lane & 15;

    v8f acc[NT];
#pragma unroll
    for (int nt = 0; nt < NT; ++nt) acc[nt] = (v8f){};

    const float* arow = A + (size_t)(m0 + l) * lda;
    for (int k = 0; k < K; k += 4) {
        int ka = k + (half << 1);                // even -> 8B aligned
        v2f a = *(const v2f*)(arow + ka);
        const float* wrow = Wp + (size_t)((k >> 1) + half) * (ncolsP * 2);
#pragma unroll
        for (int nt = 0; nt < NT; ++nt) {
            int col = (nt << 4) + l;
            v2f b = *(const v2f*)(wrow + (col << 1));
            acc[nt] = __builtin_amdgcn_wmma_f32_16x16x4_f32(
                          false, a, false, b, (short)0, acc[nt], false, false);
        }
    }
#pragma unroll
    for (int nt = 0; nt < NT; ++nt) {
        int col  = (nt << 4) + l;
        float bv = bias ? bias[col] : 0.0f;
#pragma unroll
        for (int r = 0; r < 8; ++r) {
            int row = m0 + r + (half << 3);
            C[(size_t)row * ldc + col] = acc[nt][r] + bv;
        }
    }
}

// Pack W [Krows x N] row-major into Wp [KpPairs][ncolsP][2], zero padding.
__global__ void pack_w_kernel(const float* __restrict__ w,
                              float* __restrict__ wp,
                              int Krows, int N, int KpPairs, int ncolsP)
{
    int t = blockIdx.x * blockDim.x + threadIdx.x;
    int total = KpPairs * ncolsP * 2;
    if (t >= total) return;
    int p   = t / (ncolsP * 2);
    int rem = t % (ncolsP * 2);
    int col = rem >> 1;
    int s   = rem & 1;
    int row = (p << 1) + s;
    wp[t] = (row < Krows && col < N) ? w[(size_t)row * N + col] : 0.0f;
}

// ---------------------------------------------------------------------------
// In-degree: deg[dst[e]] += 1
// ---------------------------------------------------------------------------
__global__ void degree_kernel(const int* __restrict__ dst, float* __restrict__ deg)
{
    int e = blockIdx.x * blockDim.x + threadIdx.x;
    if (e < N_EDGES) atomicAdd(&deg[dst[e]], 1.0f);
}

// Layer-1 edge scatter: agg1[dst] += tmpA[src] over GC1 dims (tmpA stride GC1P)
__global__ void scatter1_kernel(const float* __restrict__ tmpA,
                                const int* __restrict__ src,
                                const int* __restrict__ dst,
                                float* __restrict__ agg1)
{
    int t = blockIdx.x * blockDim.x + threadIdx.x;
    int e = t >> 5, lane = t & 31;
    if (e >= N_EDGES) return;
    const float* s = tmpA + (size_t)src[e] * GC1P;
    float*       d = agg1 + (size_t)dst[e] * GC1;
    for (int cidx = lane; cidx < GC1; cidx += 32)
        atomicAdd(&d[cidx], s[cidx]);
}

// Layer-2 edge scatter: agg2[dst] += h2pre[src] over DG dims (h2pre stride DGP)
__global__ void scatter2_kernel(const float* __restrict__ h2pre,
                                const int* __restrict__ src,
                                const int* __restrict__ dst,
                                float* __restrict__ agg2)
{
    int t = blockIdx.x * blockDim.x + threadIdx.x;
    int e = t >> 5, lane = t & 31;
    if (e >= N_EDGES) return;
    if (lane < DG)
        atomicAdd(&agg2[(size_t)dst[e] * DG + lane],
                  h2pre[(size_t)src[e] * DGP + lane]);
}

// h1 = relu(agg1/max(deg,1) + gc1_b)   [N_NODES x GC1], stride GC1
__global__ void h1_finalize_kernel(const float* __restrict__ agg1,
                                   const float* __restrict__ deg,
                                   const float* __restrict__ b1,
                                   float* __restrict__ h1)
{
    int t = blockIdx.x * blockDim.x + threadIdx.x;
    if (t >= N_NODES * GC1) return;
    int n = t / GC1, cidx = t % GC1;
    float v = agg1[t] / fmaxf(deg[n], 1.0f) + b1[cidx];
    h1[(size_t)n * GC1 + cidx] = fmaxf(v, 0.0f);
}

// h2 = relu(agg2/max(deg,1) + gc2_b); pool into hg_sum / gcnt by graph id
__global__ void h2_pool_kernel(const float* __restrict__ agg2,
                               const float* __restrict__ deg,
                               const float* __restrict__ b2,
                               const int* __restrict__ graph_id,
                               float* __restrict__ hg_sum,
                               float* __restrict__ gcnt)
{
    int t = blockIdx.x * blockDim.x + threadIdx.x;
    if (t >= N_NODES * DG) return;
    int n = t / DG, cidx = t % DG;
    float v = fmaxf(agg2[t] / fmaxf(deg[n], 1.0f) + b2[cidx], 0.0f);
    int g = graph_id[n];
    atomicAdd(&hg_sum[g * DG + cidx], v);
    if (cidx == 0) atomicAdd(&gcnt[g], 1.0f);
}

// Per-graph: hg = hg_sum/cnt; h_g = hg@pg_W+pg_b; h_d = desc2d@p2_W+p2_b;
// a = sigmoid(((h_g@W2) * h_d).sum())
__global__ void attn_kernel(const float* __restrict__ hg_sum,
                            const float* __restrict__ gcnt,
                            const float* __restrict__ desc2d,
                            const float* __restrict__ pg_W, const float* __restrict__ pg_b,
                            const float* __restrict__ p2_W, const float* __restrict__ p2_b,
                            const float* __restrict__ W2,
                            float* __restrict__ hg, float* __restrict__ aout)
{
    int b = blockIdx.x * blockDim.x + threadIdx.x;
    if (b >= BATCH) return;
    float cnt = fmaxf(gcnt[b], 1.0f);
    float hgl[DG];
#pragma unroll
    for (int d = 0; d < DG; ++d) {
        hgl[d] = hg_sum[b * DG + d] / cnt;
        hg[b * DG + d] = hgl[d];
    }
    float hgv[DH];
    for (int j = 0; j < DH; ++j) {
        float acc = pg_b[j];
#pragma unroll
        for (int d = 0; d < DG; ++d) acc += hgl[d] * pg_W[d * DH + j];
        hgv[j] = acc;
    }
    const float* drow = desc2d + (size_t)b * D2;
    float s = 0.0f;
    for (int j = 0; j < DH; ++j) {
        float hd = p2_b[j];
        for (int k = 0; k < D2; ++k) hd += drow[k] * p2_W[k * DH + j];
        float w = 0.0f;
        for (int i = 0; i < DH; ++i) w += hgv[i] * W2[i * DH + j];
        s += w * hd;
    }
    aout[b] = 1.0f / (1.0f + expf(-s));
}

// fusion[b, i*201+j] = hg1[b,i] * d1[b,j]; K-padded cols zeroed
__global__ void fusion_kernel(const float* __restrict__ hg,
                              const float* __restrict__ a,
                              const float* __restrict__ desc2d,
                              float* __restrict__ fusion)
{
    int t = blockIdx.x * blockDim.x + threadIdx.x;
    if (t >= BATCH * FUSEP) return;
    int b = t / FUSEP, col = t % FUSEP;
    float v = 0.0f;
    if (col < FUSE) {
        int i = col / (D2 + 1), j = col % (D2 + 1);
        float hi = (i < DG) ? hg[b * DG + i] : 1.0f;
        float dj = (j < D2) ? a[b] * desc2d[(size_t)b * D2 + j] : 1.0f;
        v = hi * dj;
    }
    fusion[t] = v;
}

// BatchNorm over batch dim (population var) + ReLU for [BATCH x H1]
__global__ void bn1_relu_kernel(const float* __restrict__ y1,
                                const float* __restrict__ g,
                                const float* __restrict__ bb,
                                float* __restrict__ z1)
{
    int j = threadIdx.x;                 // 1 block, H1 threads
    float sum = 0.0f, sq = 0.0f;
    for (int i = 0; i < BATCH; ++i) {
        float v = y1[i * H1 + j];
        sum += v; sq += v * v;
    }
    float mu  = sum / (float)BATCH;
    float var = sq / (float)BATCH - mu * mu;
    float sc  = rsqrtf(var + BN_EPS) * g[j];
    float bt  = bb[j];
    for (int i = 0; i < BATCH; ++i) {
        float v = (y1[i * H1 + j] - mu) * sc + bt;
        z1[i * H1 + j] = fmaxf(v, 0.0f);
    }
}

// y2 = z1 @ fc2_W + fc2_b   [BATCH x H2]
__global__ void fc2_kernel(const float* __restrict__ z1,
                           const float* __restrict__ w,
                           const float* __restrict__ b,
                           float* __restrict__ y2)
{
    int t = blockIdx.x * blockDim.x + threadIdx.x;
    if (t >= BATCH * H2) return;
    int r = t / H2, cidx = t % H2;
    float acc = b[cidx];
    const float* zrow = z1 + (size_t)r * H1;
    for (int k = 0; k < H1; ++k) acc += zrow[k] * w[k * H2 + cidx];
    y2[t] = acc;
}

// BN2 + ReLU + fc3  -> out [BATCH x 1]
__global__ void bn2_fc3_kernel(const float* __restrict__ y2,
                               const float* __restrict__ g,
                               const float* __restrict__ bb,
                               const float* __restrict__ fc3_W,
                               const float* __restrict__ fc3_b,
                               float* __restrict__ out)
{
    __shared__ float smu[H2], ssc[H2], sbt[H2];
    int t = threadIdx.x;                 // 1 block, BATCH threads
    if (t < H2) {
        float sum = 0.0f, sq = 0.0f;
        for (int i = 0; i < BATCH; ++i) {
            float v = y2[i * H2 + t];
            sum += v; sq += v * v;
        }
        float mu  = sum / (float)BATCH;
        float var = sq / (float)BATCH - mu * mu;
        smu[t] = mu;
        ssc[t] = rsqrtf(var + BN_EPS) * g[t];
        sbt[t] = bb[t];
    }
    __syncthreads();
    float acc = fc3_b[0];
#pragma unroll
    for (int cidx = 0; cidx < H2; ++cidx) {
        float v = (y2[t * H2 + cidx] - smu[cidx]) * ssc[cidx] + sbt[cidx];
        acc += fmaxf(v, 0.0f) * fc3_W[cidx];
    }
    out[t] = acc;
}

// ---------------------------------------------------------------------------
// Launch
// ---------------------------------------------------------------------------
static inline int cdiv(long long a, int b) { return (int)((a + b - 1) / b); }

extern "C" void kernel_launch(void* const* d_in, const int* in_sizes, int n_in,
                              void* d_out, int out_size, void* d_ws, size_t ws_size,
                              hipStream_t stream)
{
    const float* feat    = (const float*)d_in[0];
    const float* desc_2d = (const float*)d_in[1];
    // d_in[2] = desc_3d (unused by reference)
    const float* gc1_W   = (const float*)d_in[3];
    const float* gc1_b   = (const float*)d_in[4];
    const float* gc2_W   = (const float*)d_in[5];
    const float* gc2_b   = (const float*)d_in[6];
    const float* pg_W    = (const float*)d_in[7];
    const float* pg_b    = (const float*)d_in[8];
    const float* p2_W    = (const float*)d_in[9];
    const float* p2_b    = (const float*)d_in[10];
    const float* W2      = (const float*)d_in[11];
    const float* fc1_W   = (const float*)d_in[12];
    const float* fc1_b   = (const float*)d_in[13];
    const float* fc2_W   = (const float*)d_in[14];
    const float* fc2_b   = (const float*)d_in[15];
    const float* fc3_W   = (const float*)d_in[16];
    const float* fc3_b   = (const float*)d_in[17];
    const float* bn1_g   = (const float*)d_in[18];
    const float* bn1_b   = (const float*)d_in[19];
    const float* bn2_g   = (const float*)d_in[20];
    const float* bn2_b   = (const float*)d_in[21];
    const int*   src     = (const int*)d_in[22];
    const int*   dst     = (const int*)d_in[23];
    const int*   gid     = (const int*)d_in[24];
    float*       out     = (float*)d_out;

    // Packed weight sizes
    const int GC1_PAIRS = DIM_IN / 2;   // 64
    const int GC2_PAIRS = GC1 / 2;      // 50
    const int FC1_PAIRS = FUSEP / 2;    // 2112

    // Workspace layout (float offsets)
    float* ws = (float*)d_ws;
    float* tmpA   = ws;                              // 50000*112 (reused as h1, stride GC1)
    float* agg1   = tmpA  + (size_t)N_NODES * GC1P;  // 50000*100 (reused as h2pre, stride DGP)
    float* deg    = agg1  + (size_t)N_NODES * GC1;   // 50000
    float* agg2   = deg   + N_NODES;                 // 50000*20
    float* hg_sum = agg2  + (size_t)N_NODES * DG;    // 512*20
    float* gcnt   = hg_sum + BATCH * DG;             // 512
    float* hg     = gcnt  + BATCH;                   // 512*20
    float* aatt   = hg    + BATCH * DG;              // 512
    float* fusion = aatt  + BATCH;                   // 512*4224
    float* gc1Wp  = fusion + (size_t)BATCH * FUSEP;  // 64*112*2   = 14336
    float* gc2Wp  = gc1Wp + (size_t)GC1_PAIRS * GC1P * 2;  // 50*32*2 = 3200
    float* fc1Wp  = gc2Wp + (size_t)GC2_PAIRS * DGP * 2;   // 2112*128*2 = 540672
    float* y1     = fc1Wp + (size_t)FC1_PAIRS * H1 * 2;    // 512*128
    float* z1     = y1    + BATCH * H1;              // 512*128
    float* y2     = z1    + BATCH * H1;              // 512*32
    float* h1     = tmpA;                            // alias (stride GC1)
    float* h2pre  = agg1;                            // alias (stride DGP)

    // Zero accumulation buffers (required every call; graph-capture safe)
    hipMemsetAsync(agg1,   0, (size_t)N_NODES * GC1 * sizeof(float), stream);
    hipMemsetAsync(deg,    0, (size_t)N_NODES * sizeof(float), stream);
    hipMemsetAsync(agg2,   0, (size_t)N_NODES * DG * sizeof(float), stream);
    hipMemsetAsync(hg_sum, 0, (size_t)BATCH * DG * sizeof(float), stream);
    hipMemsetAsync(gcnt,   0, (size_t)BATCH * sizeof(float), stream);

    // 0) pack weight matrices (pair-interleaved, column-padded, zero-filled)
    pack_w_kernel<<<cdiv((long long)GC1_PAIRS * GC1P * 2, 256), 256, 0, stream>>>(
        gc1_W, gc1Wp, DIM_IN, GC1, GC1_PAIRS, GC1P);
    pack_w_kernel<<<cdiv((long long)GC2_PAIRS * DGP * 2, 256), 256, 0, stream>>>(
        gc2_W, gc2Wp, GC1, DG, GC2_PAIRS, DGP);
    pack_w_kernel<<<cdiv((long long)FC1_PAIRS * H1 * 2, 256), 256, 0, stream>>>(
        fc1_W, fc1Wp, FUSE, H1, FC1_PAIRS, H1);

    // 1) tmpA = feat @ gc1_W   (linear commutes with mean-aggregation)
    wmma_gemm_f32<7><<<cdiv((long long)(N_NODES / 16) * 32, 128), 128, 0, stream>>>(
        feat, DIM_IN, gc1Wp, tmpA, GC1P, nullptr, N_NODES / 16, DIM_IN);
    // 2) in-degree
    degree_kernel<<<cdiv(N_EDGES, 256), 256, 0, stream>>>(dst, deg);
    // 3) agg1[dst] += tmpA[src]
    scatter1_kernel<<<cdiv((long long)N_EDGES * 32, 256), 256, 0, stream>>>(
        tmpA, src, dst, agg1);
    // 4) h1 = relu(agg1/deg + b1)  (writes over tmpA region, stride GC1)
    h1_finalize_kernel<<<cdiv((long long)N_NODES * GC1, 256), 256, 0, stream>>>(
        agg1, deg, gc1_b, h1);
    // 5) h2pre = h1 @ gc2_W  (writes over agg1 region, stride DGP)
    wmma_gemm_f32<2><<<cdiv((long long)(N_NODES / 16) * 32, 128), 128, 0, stream>>>(
        h1, GC1, gc2Wp, h2pre, DGP, nullptr, N_NODES / 16, GC1);
    // 6) agg2[dst] += h2pre[src]
    scatter2_kernel<<<cdiv((long long)N_EDGES * 32, 256), 256, 0, stream>>>(
        h2pre, src, dst, agg2);
    // 7) h2 = relu(agg2/deg + b2), pooled by graph id
    h2_pool_kernel<<<cdiv((long long)N_NODES * DG, 256), 256, 0, stream>>>(
        agg2, deg, gc2_b, gid, hg_sum, gcnt);
    // 8) attention scalar per graph
    attn_kernel<<<cdiv(BATCH, 256), 256, 0, stream>>>(
        hg_sum, gcnt, desc_2d, pg_W, pg_b, p2_W, p2_b, W2, hg, aatt);
    // 9) fusion outer product (K-padded cols zeroed each call)
    fusion_kernel<<<cdiv((long long)BATCH * FUSEP, 256), 256, 0, stream>>>(
        hg, aatt, desc_2d, fusion);
    // 10) y1 = fusion @ fc1_W + fc1_b
    wmma_gemm_f32<8><<<cdiv((long long)(BATCH / 16) * 32, 128), 128, 0, stream>>>(
        fusion, FUSEP, fc1Wp, y1, H1, fc1_b, BATCH / 16, FUSEP);
    // 11) BN1 + ReLU
    bn1_relu_kernel<<<1, H1, 0, stream>>>(y1, bn1_g, bn1_b, z1);
    // 12) fc2
    fc2_kernel<<<cdiv((long long)BATCH * H2, 256), 256, 0, stream>>>(
        z1, fc2_W, fc2_b, y2);
    // 13) BN2 + ReLU + fc3 -> out
    bn2_fc3_kernel<<<1, BATCH, 0, stream>>>(y2, bn2_g, bn2_b, fc3_W, fc3_b, out);
}